// GATr_to_e_78950088835243
// MI455X (gfx1250) — compile-verified
//
#include <hip/hip_runtime.h>
#include <math.h>

typedef __attribute__((ext_vector_type(2))) float v2f;
typedef __attribute__((ext_vector_type(4))) float v4f;
typedef __attribute__((ext_vector_type(8))) float v8f;

#define E_HID 64
#define IN_DIM 192
#define NEG_SLOPE 0.01f
#define ENC_NEG_INF 0x007FFFFFu  // enc(-inf)

// Order-preserving float -> uint mapping (monotone), for atomic max on floats.
__device__ __forceinline__ unsigned enc_f(float f) {
  unsigned u = __float_as_uint(f);
  return (u & 0x80000000u) ? ~u : (u | 0x80000000u);
}
__device__ __forceinline__ float dec_f(unsigned u) {
  unsigned v = (u & 0x80000000u) ? (u & 0x7fffffffu) : ~u;
  return __uint_as_float(v);
}

// ---------------------------------------------------------------------------
// k_init: per destination node n:
//   nd[n] = dot(x_e[n], a_node); m_enc[n] = enc(-inf); s[n] = 0; agg[n][:] = 0
// 16 threads per node (float4 each), width-16 shuffle reduction (wave32-safe:
// node groups align to 16-lane halves).
// ---------------------------------------------------------------------------
__global__ __launch_bounds__(256) void k_init(
    const float* __restrict__ x_e, const float* __restrict__ a_node,
    float* __restrict__ nd, unsigned* __restrict__ m_enc,
    float* __restrict__ s, float* __restrict__ agg, int n_nodes) {
  int node = blockIdx.x * 16 + (threadIdx.x >> 4);
  int t = threadIdx.x & 15;
  if (node >= n_nodes) return;
  v4f xv = *(const v4f*)(x_e + (size_t)node * E_HID + t * 4);
  v4f av = *(const v4f*)(a_node + t * 4);
  float p = xv[0] * av[0] + xv[1] * av[1] + xv[2] * av[2] + xv[3] * av[3];
  for (int off = 8; off; off >>= 1) p += __shfl_xor(p, off, 16);
  v4f z = {0.f, 0.f, 0.f, 0.f};
  *(v4f*)(agg + (size_t)node * E_HID + t * 4) = z;
  if (t == 0) {
    nd[node] = p;
    m_enc[node] = ENC_NEG_INF;
    s[node] = 0.f;
  }
}

// ---------------------------------------------------------------------------
// k_gemm: e_r = x_r @ W^T + bias  via V_WMMA_F32_16X16X4_F32 (fp32-exact).
// One wave per 16-row tile of x_r; N=64 -> 4 column tiles; K=192 -> 48 steps.
// Fused epilogue: bias add + rel_dot[e] = e_r[e] . a_rel (cross-lane reduce).
// A frag (16x4 f32): lane<16 -> row=l16, K={kb,kb+1}; lane>=16 -> K={kb+2,kb+3}
// B frag (4x16 f32): lane -> col=l16, same K split. C: VGPR v -> row v+8*half.
// `tile` is forced into SGPRs via readfirstlane so the tile bound checks and
// the full/tail epilogue selection are scalar branches (no exec-mask churn).
// ---------------------------------------------------------------------------
__global__ __launch_bounds__(256) void k_gemm(
    const float* __restrict__ x_r, const float* __restrict__ W,
    const float* __restrict__ bias, const float* __restrict__ a_rel,
    float* __restrict__ e_r, float* __restrict__ rel_dot,
    int n_edges, int n_tiles) {
  __shared__ float sW[E_HID * IN_DIM];  // 48 KB
  __shared__ float sBias[E_HID];
  __shared__ float sArel[E_HID];
  for (int i = threadIdx.x; i < E_HID * IN_DIM; i += 256) sW[i] = W[i];
  if (threadIdx.x < E_HID) {
    sBias[threadIdx.x] = bias[threadIdx.x];
    sArel[threadIdx.x] = a_rel[threadIdx.x];
  }
  __syncthreads();

  // Wave id is uniform per wave; make it provably uniform (SGPR).
  const int wave = __builtin_amdgcn_readfirstlane(threadIdx.x) >> 5;
  const int lane = threadIdx.x & 31;
  const int l16 = lane & 15;
  const int hf = lane >> 4;
  const int tile = blockIdx.x * 8 + wave;  // scalar
  if (tile >= n_tiles) return;  // scalar branch (EXEC stays all-ones)

  int arow = tile * 16 + l16;
  if (arow >= n_edges) arow = n_edges - 1;  // clamp tail (stores are guarded)
  const float* ap = x_r + (size_t)arow * IN_DIM + 2 * hf;
  const float* wp0 = sW + l16 * IN_DIM + 2 * hf;

  v8f acc[4] = {};
  for (int kb = 0; kb < IN_DIM; kb += 4) {
    v2f A = *(const v2f*)(ap + kb);
#pragma unroll
    for (int t = 0; t < 4; ++t) {
      v2f B = *(const v2f*)(wp0 + kb + t * 16 * IN_DIM);
      acc[t] = __builtin_amdgcn_wmma_f32_16x16x4_f32(
          false, A, false, B, (short)0, acc[t], false, false);
    }
  }

  // Epilogue: bias + store e_r + fused rel_dot reduction.
  const bool full_tile = (tile * 16 + 16 <= n_edges);  // scalar predicate
  // Per-lane base: row (tile*16 + 8*hf), column l16.
  float* erp = e_r + ((size_t)tile * 16 + 8 * hf) * E_HID + l16;

  float bs[4], wt[4];
#pragma unroll
  for (int t = 0; t < 4; ++t) {
    bs[t] = sBias[t * 16 + l16];
    wt[t] = sArel[t * 16 + l16];
  }

  float pv[8];
#pragma unroll
  for (int v = 0; v < 8; ++v) pv[v] = 0.f;

  if (full_tile) {
#pragma unroll
    for (int v = 0; v < 8; ++v) {
#pragma unroll
      for (int t = 0; t < 4; ++t) {
        float val = acc[t][v] + bs[t];
        erp[(size_t)v * E_HID + t * 16] = val;  // branch-free, coalesced
        pv[v] += val * wt[t];
      }
    }
  } else {
#pragma unroll
    for (int v = 0; v < 8; ++v) {
      int r = tile * 16 + 8 * hf + v;
#pragma unroll
      for (int t = 0; t < 4; ++t) {
        float val = acc[t][v] + bs[t];
        if (r < n_edges) erp[(size_t)v * E_HID + t * 16] = val;
        pv[v] += val * wt[t];
      }
    }
  }

#pragma unroll
  for (int v = 0; v < 8; ++v) {
    float p = pv[v];
    for (int off = 8; off; off >>= 1) p += __shfl_xor(p, off, 16);
    pv[v] = p;
  }
  if (l16 == 0) {
    if (full_tile) {
#pragma unroll
      for (int v = 0; v < 8; ++v) rel_dot[tile * 16 + 8 * hf + v] = pv[v];
    } else {
#pragma unroll
      for (int v = 0; v < 8; ++v) {
        int r = tile * 16 + 8 * hf + v;
        if (r < n_edges) rel_dot[r] = pv[v];
      }
    }
  }
}

// ---------------------------------------------------------------------------
// k_logits: lr[e] = leaky_relu(nd[idx[e]] + rel_dot[e]); atomic max per seg.
// ---------------------------------------------------------------------------
__global__ __launch_bounds__(256) void k_logits(
    const int* __restrict__ idx, const float* __restrict__ nd,
    const float* __restrict__ rel_dot, float* __restrict__ lr,
    unsigned* __restrict__ m_enc, int n_edges) {
  int e = blockIdx.x * 256 + threadIdx.x;
  if (e >= n_edges) return;
  int d = idx[e];
  float lg = nd[d] + rel_dot[e];
  float l = lg > 0.f ? lg : NEG_SLOPE * lg;
  lr[e] = l;
  atomicMax(m_enc + d, enc_f(l));
}

// ---------------------------------------------------------------------------
// k_exp: ex[e] = exp(lr[e] - m[idx[e]]) (stored over lr); s[idx[e]] += ex.
// ---------------------------------------------------------------------------
__global__ __launch_bounds__(256) void k_exp(
    const int* __restrict__ idx, const unsigned* __restrict__ m_enc,
    float* __restrict__ lr, float* __restrict__ s, int n_edges) {
  int e = blockIdx.x * 256 + threadIdx.x;
  if (e >= n_edges) return;
  int d = idx[e];
  float ex = expf(lr[e] - dec_f(m_enc[d]));
  lr[e] = ex;
  atomicAdd(s + d, ex);
}

// ---------------------------------------------------------------------------
// k_agg: agg[idx[e]] += alpha_e * e_r[rel_size[e]]; 16 threads/edge, float4.
// agg is 25.6 MB -> L2-resident, so the float atomics stay on-chip.
// ---------------------------------------------------------------------------
__global__ __launch_bounds__(256) void k_agg(
    const int* __restrict__ idx, const int* __restrict__ rel,
    const float* __restrict__ ex, const float* __restrict__ s,
    const float* __restrict__ e_r, float* __restrict__ agg, int n_edges) {
  int e = blockIdx.x * 16 + (threadIdx.x >> 4);
  int t = threadIdx.x & 15;
  if (e >= n_edges) return;
  int d = idx[e];
  float alpha = ex[e] / (s[d] + 1e-16f);
  int r = rel[e];
  v4f v = *(const v4f*)(e_r + (size_t)r * E_HID + t * 4);
  float* a = agg + (size_t)d * E_HID + t * 4;
  atomicAdd(a + 0, alpha * v[0]);
  atomicAdd(a + 1, alpha * v[1]);
  atomicAdd(a + 2, alpha * v[2]);
  atomicAdd(a + 3, alpha * v[3]);
}

// ---------------------------------------------------------------------------
// k_update: x_e += relu(agg), elementwise (in-place on d_out).
// ---------------------------------------------------------------------------
__global__ __launch_bounds__(256) void k_update(
    float* __restrict__ x_e, const float* __restrict__ agg, long long n) {
  long long i = (long long)blockIdx.x * 256 + threadIdx.x;
  if (i >= n) return;
  float g = agg[i];
  x_e[i] += g > 0.f ? g : 0.f;
}

// ---------------------------------------------------------------------------
extern "C" void kernel_launch(void* const* d_in, const int* in_sizes, int n_in,
                              void* d_out, int out_size, void* d_ws,
                              size_t ws_size, hipStream_t stream) {
  const float* x_e_in = (const float*)d_in[0];
  const float* x_r = (const float*)d_in[1];
  const int* edge_index = (const int*)d_in[2];
  const int* rel_size = (const int*)d_in[3];
  const float* Wr = (const float*)d_in[4];
  const float* br = (const float*)d_in[5];
  const float* Wr1 = (const float*)d_in[6];
  const float* br1 = (const float*)d_in[7];
  const float* Wr2 = (const float*)d_in[8];
  const float* br2 = (const float*)d_in[9];
  const float* ah = (const float*)d_in[10];
  const float* ah1 = (const float*)d_in[11];
  const float* at = (const float*)d_in[12];
  const float* ar1 = (const float*)d_in[13];
  const float* ar2 = (const float*)d_in[14];
  const float* ar3 = (const float*)d_in[15];

  const int n_nodes = in_sizes[0] / E_HID;
  const int n_edges = in_sizes[3];
  const int* hsrc = edge_index;            // edge_index[0]
  const int* tsrc = edge_index + n_edges;  // edge_index[1]

  float* x_e = (float*)d_out;
  hipMemcpyAsync(x_e, x_e_in, (size_t)n_nodes * E_HID * sizeof(float),
                 hipMemcpyDeviceToDevice, stream);

  // Workspace carve-up (~159 MB total).
  char* w = (char*)d_ws;
  float* e_r = (float*)w;      w += (size_t)n_edges * E_HID * sizeof(float);
  float* agg = (float*)w;      w += (size_t)n_nodes * E_HID * sizeof(float);
  float* rel_dot = (float*)w;  w += (size_t)n_edges * sizeof(float);
  float* lr = (float*)w;       w += (size_t)n_edges * sizeof(float);
  float* nd = (float*)w;       w += (size_t)n_nodes * sizeof(float);
  unsigned* m_enc = (unsigned*)w; w += (size_t)n_nodes * sizeof(unsigned);
  float* s = (float*)w;        w += (size_t)n_nodes * sizeof(float);

  const int n_tiles = (n_edges + 15) / 16;
  const int g_node16 = (n_nodes + 15) / 16;
  const int g_gemm = (n_tiles + 7) / 8;
  const int g_edge = (n_edges + 255) / 256;
  const int g_edge16 = (n_edges + 15) / 16;
  const long long n_xe = (long long)n_nodes * E_HID;
  const int g_upd = (int)((n_xe + 255) / 256);

  struct Blk {
    const int* idx; const float* W; const float* b;
    const float* an; const float* ar;
  };
  Blk blk[3] = {
      {hsrc, Wr, br, ah, ar1},
      {tsrc, Wr1, br1, at, ar2},
      {hsrc, Wr2, br2, ah1, ar3},
  };

  for (int b = 0; b < 3; ++b) {
    k_init<<<g_node16, 256, 0, stream>>>(x_e, blk[b].an, nd, m_enc, s, agg,
                                         n_nodes);
    k_gemm<<<g_gemm, 256, 0, stream>>>(x_r, blk[b].W, blk[b].b, blk[b].ar,
                                       e_r, rel_dot, n_edges, n_tiles);
    k_logits<<<g_edge, 256, 0, stream>>>(blk[b].idx, nd, rel_dot, lr, m_enc,
                                         n_edges);
    k_exp<<<g_edge, 256, 0, stream>>>(blk[b].idx, m_enc, lr, s, n_edges);
    k_agg<<<g_edge16, 256, 0, stream>>>(blk[b].idx, rel_size, lr, s, e_r, agg,
                                        n_edges);
    k_update<<<g_upd, 256, 0, stream>>>(x_e, agg, n_xe);
  }
}